// GNNAnomalyDetector_26508538150912
// MI455X (gfx1250) — compile-verified
//
#include <hip/hip_runtime.h>

typedef __attribute__((ext_vector_type(2))) float v2f;
typedef __attribute__((ext_vector_type(8))) float v8f;

#define N_NODES 50000
#define N_EDGES 800000
#define IN_CH 64
#define HID 128

// ---------------------------------------------------------------------------
// degree: deg[dst[e]] += 1.0f
// ---------------------------------------------------------------------------
__global__ void deg_kernel(const long long* __restrict__ dst,
                           float* __restrict__ deg, int nE) {
  int e = blockIdx.x * blockDim.x + threadIdx.x;
  if (e < nE) atomicAdd(&deg[(int)dst[e]], 1.0f);
}

// ---------------------------------------------------------------------------
// scatter-add: agg[dst[e]] += feat[src[e]], D channels, float4-vectorized
// ---------------------------------------------------------------------------
template <int D>
__global__ void scatter_add_kernel(const long long* __restrict__ src,
                                   const long long* __restrict__ dst,
                                   const float* __restrict__ feat,
                                   float* __restrict__ agg, int nwork) {
  int idx = blockIdx.x * blockDim.x + threadIdx.x;
  if (idx >= nwork) return;
  constexpr int C4 = D / 4;
  int e = idx / C4;
  int c = (idx - e * C4) * 4;
  int s = (int)src[e];
  int d = (int)dst[e];
  const float4 v = *(const float4*)(feat + (size_t)s * D + c);
  float* out = agg + (size_t)d * D + c;
  atomicAdd(out + 0, v.x);
  atomicAdd(out + 1, v.y);
  atomicAdd(out + 2, v.z);
  atomicAdd(out + 3, v.w);
}

// ---------------------------------------------------------------------------
// Fused dual-GEMM with f32 WMMA (16x16x4):
//   out[m][n] = (A1[m]/max(deg[m],1)) . W1[n] + bias[n] + A2[m] . W2[n]
// One 16-row M tile per block; one 16-col N tile per wave (blockDim = N/16*32).
// deg==nullptr -> no mean scaling; A2==nullptr -> single GEMM; relu optional.
// W* are [N][K] row-major (PyG weight layout), so B = W^T loads are contiguous.
// ---------------------------------------------------------------------------
template <int K>
__global__ __launch_bounds__(256) void gemm_dual_wmma(
    const float* __restrict__ A1, const float* __restrict__ deg,
    const float* __restrict__ W1,
    const float* __restrict__ A2, const float* __restrict__ W2,
    const float* __restrict__ bias, float* __restrict__ out,
    int N, int relu) {
  __shared__ float sA1[16 * K];
  __shared__ float sA2[16 * K];
  const int m0 = blockIdx.x * 16;

  // Stage the 16xK A tiles in LDS; fuse mean = agg / max(deg,1) into the load.
  for (int i = threadIdx.x; i < 16 * K; i += blockDim.x) {
    int r = i / K;
    int k = i - r * K;
    float s = 1.0f;
    if (deg) s = 1.0f / fmaxf(deg[m0 + r], 1.0f);
    sA1[i] = A1[(size_t)(m0 + r) * K + k] * s;
    if (A2) sA2[i] = A2[(size_t)(m0 + r) * K + k];
  }
  __syncthreads();

  const int lane = threadIdx.x & 31;
  const int wave = threadIdx.x >> 5;
  const int n0   = wave * 16;
  const int mr   = lane & 15;        // A row within tile / C-D column
  const int kg   = (lane >> 4) * 2;  // K sub-offset (lanes 16-31 hold K=2,3)
  const int nc   = n0 + mr;          // global output column

  v8f acc = {};
  for (int k0 = 0; k0 < K; k0 += 4) {
    v2f a = *(const v2f*)&sA1[mr * K + k0 + kg];
    v2f b = *(const v2f*)&W1[(size_t)nc * K + k0 + kg];
    acc = __builtin_amdgcn_wmma_f32_16x16x4_f32(false, a, false, b, (short)0,
                                                acc, false, false);
  }
  if (A2) {
    for (int k0 = 0; k0 < K; k0 += 4) {
      v2f a = *(const v2f*)&sA2[mr * K + k0 + kg];
      v2f b = *(const v2f*)&W2[(size_t)nc * K + k0 + kg];
      acc = __builtin_amdgcn_wmma_f32_16x16x4_f32(false, a, false, b, (short)0,
                                                  acc, false, false);
    }
  }

  // C/D layout: VGPR v -> row m0+v (lanes 0-15) / m0+v+8 (lanes 16-31), col nc
  const float bv = bias[nc];
  const int mb = m0 + ((lane >> 4) << 3);
#pragma unroll
  for (int v = 0; v < 8; v++) {
    float r = acc[v] + bv;
    if (relu) r = fmaxf(r, 0.0f);
    out[(size_t)(mb + v) * N + nc] = r;
  }
}

// ---------------------------------------------------------------------------
// Launch: deg -> scatter1 -> SAGE1(relu) -> scatter2 -> SAGE2 -> recon
// ---------------------------------------------------------------------------
extern "C" void kernel_launch(void* const* d_in, const int* in_sizes, int n_in,
                              void* d_out, int out_size, void* d_ws,
                              size_t ws_size, hipStream_t stream) {
  (void)in_sizes; (void)n_in; (void)out_size; (void)ws_size;

  const float*     x    = (const float*)d_in[0];
  const long long* edge = (const long long*)d_in[1];  // int64 [2, E]
  const float*     W1l  = (const float*)d_in[2];
  const float*     b1l  = (const float*)d_in[3];
  const float*     W1r  = (const float*)d_in[4];
  const float*     W2l  = (const float*)d_in[5];
  const float*     b2l  = (const float*)d_in[6];
  const float*     W2r  = (const float*)d_in[7];
  const float*     Wrec = (const float*)d_in[8];
  const float*     brec = (const float*)d_in[9];

  const long long* src = edge;            // edge_index[0]
  const long long* dst = edge + N_EDGES;  // edge_index[1]

  // Workspace layout (16B-aligned offsets):
  char*  ws   = (char*)d_ws;
  float* deg  = (float*)(ws);             // N floats           (0.2 MB)
  float* agg1 = (float*)(ws + 200192);    // N*64 floats        (12.8 MB)
  float* h    = (float*)(ws + 13000192);  // N*128 floats       (25.6 MB)
  float* agg2 = (float*)(ws + 38600192);  // N*128 floats       (25.6 MB)

  float* z    = (float*)d_out;                              // N*128
  float* xrec = (float*)d_out + (size_t)N_NODES * HID;      // N*64

  hipMemsetAsync(deg, 0, (size_t)N_NODES * sizeof(float), stream);
  hipMemsetAsync(agg1, 0, (size_t)N_NODES * IN_CH * sizeof(float), stream);
  hipMemsetAsync(agg2, 0, (size_t)N_NODES * HID * sizeof(float), stream);

  deg_kernel<<<(N_EDGES + 255) / 256, 256, 0, stream>>>(dst, deg, N_EDGES);

  {  // layer 1 aggregation (64 ch)
    int nwork = N_EDGES * (IN_CH / 4);
    scatter_add_kernel<IN_CH>
        <<<(nwork + 255) / 256, 256, 0, stream>>>(src, dst, x, agg1, nwork);
  }
  // h = relu(mean1 @ W1l^T + b1l + x @ W1r^T)
  gemm_dual_wmma<IN_CH><<<N_NODES / 16, 256, 0, stream>>>(
      agg1, deg, W1l, x, W1r, b1l, h, HID, 1);

  {  // layer 2 aggregation (128 ch)
    int nwork = N_EDGES * (HID / 4);
    scatter_add_kernel<HID>
        <<<(nwork + 255) / 256, 256, 0, stream>>>(src, dst, h, agg2, nwork);
  }
  // z = mean2 @ W2l^T + b2l + h @ W2r^T   (no activation)
  gemm_dual_wmma<HID><<<N_NODES / 16, 256, 0, stream>>>(
      agg2, deg, W2l, h, W2r, b2l, z, HID, 0);

  // x_recon = z @ Wrec^T + brec   (N = 64 -> 4 waves / 128 threads)
  gemm_dual_wmma<HID><<<N_NODES / 16, 128, 0, stream>>>(
      z, nullptr, Wrec, nullptr, nullptr, brec, xrec, IN_CH, 0);
}